// RandomProjectionQuantizer_28724741275697
// MI455X (gfx1250) — compile-verified
//
#include <hip/hip_runtime.h>
#include <hip/hip_bf16.h>

typedef __attribute__((ext_vector_type(2))) float v2f;
typedef __attribute__((ext_vector_type(4))) float v4f;
typedef __attribute__((ext_vector_type(8))) float v8f;
typedef __attribute__((ext_vector_type(4))) unsigned int u32x4;
typedef __attribute__((ext_vector_type(4))) int i32x4;
typedef __attribute__((ext_vector_type(8))) int i32x8;

#define D_IN   512
#define DC     16
#define KCODES 8192
#define WAVES  8
#define TOK_PER_BLOCK 128
#define W_STRIDE 516            // padded W row stride (dwords): 516%64=4 -> conflict-free
#define HT_STRIDE 20            // padded h-transpose row stride
#define CB_STAGE 256            // codes staged per TDM transfer
#define CB_ROW 20               // LDS codebook row stride: 16 data + 4 pad dwords (TDM pad)
#define N_STAGES (KCODES / CB_STAGE)

// ---- Tensor Data Mover: 1D f32 tile load, optional LDS padding (D# per ISA 08 §8) ----
// This toolchain exposes the 6-arg builtin:
//   (uint32x4 group0, int32x8 group1, int32x4 group2, int32x4 group3, int32x8 extra, i32 cpol)
static __device__ __forceinline__ void tdm_load_f32_1d(unsigned int lds_byte,
                                                       const float* gsrc,
                                                       unsigned int nelem,
                                                       bool pad /*4 dw per 16 dw*/) {
  unsigned long long ga = (unsigned long long)(uintptr_t)gsrc;
  u32x4 g0;
  g0[0] = 1u;                                          // count=1, user mode, no gather
  g0[1] = lds_byte;                                    // D#.lds_addr
  g0[2] = (unsigned int)ga;                            // global_addr[31:0]
  g0[3] = (unsigned int)((ga >> 32) & 0x01FFFFFFu) | (2u << 30);  // addr[56:32] | type=2
  i32x8 g1;
  unsigned int d0 = (2u << 16);                        // data_size = 4B
  if (pad) d0 |= (1u << 20) | (3u << 22) | (3u << 25); // pad_en, interval=16dw, amount=4dw
  g1[0] = (int)d0;                                     // wg_mask=0 (not in cluster)
  g1[1] = (int)((nelem & 0xFFFFu) << 16);              // atomic_barrier=0 | tensor_dim0.lo
  g1[2] = (int)(((nelem >> 16) & 0xFFFFu) | (1u << 16)); // tensor_dim0.hi | tensor_dim1=1
  g1[3] = (int)((nelem & 0xFFFFu) << 16);              // tensor_dim1.hi=0 | tile_dim0
  g1[4] = 0;                                           // tile_dim1=0, tile_dim2=0 (1D)
  g1[5] = (int)nelem;                                  // tensor_dim0_stride
  g1[6] = 0; g1[7] = 0;
  i32x4 z4 = {};
  i32x8 z8 = {};
  __builtin_amdgcn_tensor_load_to_lds(g0, g1, z4, z4, z8, 0);
}

static __device__ __forceinline__ unsigned int lds_off(const void* p) {
  return (unsigned int)(uintptr_t)p;   // LDS offset lives in low 32 bits of generic addr
}

__global__ __launch_bounds__(256)
void rpq_kernel(const float* __restrict__ x,
                const float* __restrict__ W,
                const float* __restrict__ cb,
                int* __restrict__ out) {
  __shared__ float Wl[DC * W_STRIDE];                  // 33.0 KB padded proj weight
  __shared__ float cbL[2 * CB_STAGE * CB_ROW];         // 40 KB double-buffered codebook
  __shared__ float c2L[2 * CB_STAGE];                  // 2 KB per-stage |c|^2
  __shared__ float hT[WAVES][16 * HT_STRIDE];          // 10 KB h transpose

  const int tid  = threadIdx.x;
  const int wave = tid >> 5;
  const int lane = tid & 31;
  const int lm   = lane & 15;      // token/code row within 16-tile
  const int lh   = lane >> 4;      // half select (K pairing in f32 WMMA A/B layout)

  // ---- stage W into padded LDS (vmem pipe; one-shot) ----
  for (int i = tid; i < DC * D_IN / 4; i += 256) {
    v4f v = ((const v4f*)W)[i];
    int e = i << 2;
    *(v4f*)(&Wl[(e >> 9) * W_STRIDE + (e & 511)]) = v;
  }
  // kick TDM for codebook stage 0 early: DMA overlaps the whole projection phase
  if (wave == 0) tdm_load_f32_1d(lds_off(&cbL[0]), cb, CB_STAGE * DC, true);
  __syncthreads();                                     // W ready

  const int tokenBase = blockIdx.x * TOK_PER_BLOCK + wave * 16;

  // ---- projection: h(16x16) = x_tile(16x512) @ W^T(512x16), chained f32 WMMA ----
  v8f acc = {};
  {
    const float* xrow = x + (size_t)(tokenBase + lm) * D_IN + (lh << 1);
    const float* wrow = Wl + lm * W_STRIDE + (lh << 1);
    #pragma unroll 4
    for (int kk = 0; kk < D_IN; kk += 4) {
      v2f a = *(const v2f*)(xrow + kk);
      v2f b = *(const v2f*)(wrow + kk);
      acc = __builtin_amdgcn_wmma_f32_16x16x4_f32(false, a, false, b,
                                                  (short)0, acc, false, false);
    }
  }

  // ---- LayerNorm over Dc=16 per token, pre-scaled by -2 for the distance GEMM ----
  // masks 1/2/4/8 never cross the 16-lane group -> plain width-32 xor shuffles
  float hs[8];
  #pragma unroll
  for (int r = 0; r < 8; ++r) {
    float v = acc[r];
    float s = v, q = v * v;
    #pragma unroll
    for (int m = 8; m >= 1; m >>= 1) {
      s += __shfl_xor(s, m, 32);
      q += __shfl_xor(q, m, 32);
    }
    float mu  = s * 0.0625f;
    float var = q * 0.0625f - mu * mu;
    float inv = rsqrtf(var + 1e-5f);
    hs[r] = -2.0f * (v - mu) * inv;
  }

  // ---- transpose h: C-layout -> B-layout via padded LDS ----
  {
    float* hw = hT[wave];
    #pragma unroll
    for (int r = 0; r < 8; ++r)
      hw[(r + (lh << 3)) * HT_STRIDE + lm] = hs[r];
  }
  if (wave == 0) __builtin_amdgcn_s_wait_tensorcnt((short)0);  // stage 0 DMA done
  __syncthreads();                                             // hT + cb stage 0 visible

  // B fragments of h' (20 x 16 tokens); K=16 row is the constant 1.0 (c^2 hook)
  v2f bs[5];
  {
    const float* hw = hT[wave] + lm * HT_STRIDE + (lh << 1);
    #pragma unroll
    for (int s = 0; s < 4; ++s) bs[s] = *(const v2f*)(hw + 4 * s);
    v2f b4; b4[0] = (lh == 0) ? 1.0f : 0.0f; b4[1] = 0.0f;
    bs[4] = b4;
  }

  // ---- codebook scan: TDM double-buffered stages of 256 codes ----
  float minv = 3.4e38f;
  int   mini = 0;
  for (int s = 0; s < N_STAGES; ++s) {
    const int cur = s & 1;
    if (wave == 0 && s + 1 < N_STAGES)                 // prefetch next stage via TDM
      tdm_load_f32_1d(lds_off(&cbL[(cur ^ 1) * CB_STAGE * CB_ROW]),
                      cb + (size_t)(s + 1) * CB_STAGE * DC, CB_STAGE * DC, true);

    // per-stage c^2: one thread per code, computed once per block
    {
      const float* row = &cbL[cur * CB_STAGE * CB_ROW + tid * CB_ROW];
      float sum = 0.f;
      #pragma unroll
      for (int j = 0; j < DC; ++j) sum += row[j] * row[j];
      c2L[cur * CB_STAGE + tid] = sum;
    }
    __syncthreads();                                   // c2 ready (cb[cur] already ready)

    const float* cbl = &cbL[cur * CB_STAGE * CB_ROW];
    const float* c2p = &c2L[cur * CB_STAGE];
    #pragma unroll 2
    for (int c0 = 0; c0 < CB_STAGE; c0 += 16) {
      const float* crow = cbl + (c0 + lm) * CB_ROW + (lh << 1);
      v2f a0 = *(const v2f*)(crow + 0);
      v2f a1 = *(const v2f*)(crow + 4);
      v2f a2 = *(const v2f*)(crow + 8);
      v2f a3 = *(const v2f*)(crow + 12);
      float c2v = c2p[c0 + lm];                        // broadcast-friendly read
      v2f a4; a4[0] = (lh == 0) ? c2v : 0.0f; a4[1] = 0.0f;

      v8f d = {};
      d = __builtin_amdgcn_wmma_f32_16x16x4_f32(false, a0, false, bs[0], (short)0, d, false, false);
      d = __builtin_amdgcn_wmma_f32_16x16x4_f32(false, a1, false, bs[1], (short)0, d, false, false);
      d = __builtin_amdgcn_wmma_f32_16x16x4_f32(false, a2, false, bs[2], (short)0, d, false, false);
      d = __builtin_amdgcn_wmma_f32_16x16x4_f32(false, a3, false, bs[3], (short)0, d, false, false);
      d = __builtin_amdgcn_wmma_f32_16x16x4_f32(false, a4, false, bs[4], (short)0, d, false, false);

      // D: d[r] -> code = base + r (+8 for upper half), token = lane%16
      #pragma unroll
      for (int r = 0; r < 8; ++r) {
        float dv = d[r];
        int code = s * CB_STAGE + c0 + r + (lh << 3);
        if (dv < minv) { minv = dv; mini = code; }
      }
    }
    if (wave == 0) __builtin_amdgcn_s_wait_tensorcnt((short)0);  // next stage DMA done
    __syncthreads();                                   // publish next buf; cur reusable
  }

  // combine lane pair (t, t+16); lower code index wins ties
  {
    float ov = __shfl_xor(minv, 16, 32);
    int   oi = __shfl_xor(mini, 16, 32);
    if (ov < minv || (ov == minv && oi < mini)) { minv = ov; mini = oi; }
  }
  if (lh == 0) out[tokenBase + lm] = mini;
}

extern "C" void kernel_launch(void* const* d_in, const int* in_sizes, int n_in,
                              void* d_out, int out_size, void* d_ws, size_t ws_size,
                              hipStream_t stream) {
  (void)n_in; (void)d_ws; (void)ws_size; (void)out_size;
  const float* x  = (const float*)d_in[0];
  const float* W  = (const float*)d_in[1];
  const float* cb = (const float*)d_in[2];
  int* out = (int*)d_out;
  const int ntok   = in_sizes[0] / D_IN;                 // 16384
  const int blocks = (ntok + TOK_PER_BLOCK - 1) / TOK_PER_BLOCK;
  rpq_kernel<<<blocks, 256, 0, stream>>>(x, W, cb, out);
}